// tree_gru_onehot_revised_60971355734162
// MI455X (gfx1250) — compile-verified
//
#include <hip/hip_runtime.h>
#include <math.h>
#include <utility>

// ---------------- problem constants (match reference) ----------------
constexpr int N_NODES = 20000;
constexpr int N_EDGES = 320000;
constexpr int VOCAB   = 100;
constexpr int H       = 128;
constexpr int HEADS   = 3;
constexpr int LAYERS  = 2;
constexpr int BATCH   = 200;
constexpr int T_SEQ   = LAYERS + 1;   // readout sequence length
constexpr float NEG_SLOPE = 0.2f;
constexpr float LN_EPS    = 1e-5f;

typedef __attribute__((ext_vector_type(2))) float v2f;
typedef __attribute__((ext_vector_type(8))) float v8f;

// ---------------- WMMA f32 GEMM with async-LDS double-buffered W panel ----------------
// C[M,Nout] = A[M,K] @ W[K,Nout] + bias.
// Block = 4 waves. Wave w computes M-tile (blockIdx.y*4+w) x 64 columns (4 accumulators).
// The 32x64 W panel for the block's columns is staged in LDS with
// global_load_async_to_lds_b128 (ASYNCcnt) and double-buffered.
// Requires K%32==0, Nout%64==0 (true for all call sites: K in {128,384}, Nout in {128,384}).
constexpr int KC = 32;   // k-chunk staged per LDS buffer

__global__ void k_gemm_wmma(const float* __restrict__ A, const float* __restrict__ W,
                            const float* __restrict__ bias, float* __restrict__ C,
                            int M, int K, int Nout) {
  __shared__ float wpanel[2][KC * 64];
  const int tid  = threadIdx.x;
  const int lane = tid & 31;
  const int wave = tid >> 5;
  const int nTilesM = M >> 4;
  int tileM = blockIdx.y * 4 + wave;
  const bool active = tileM < nTilesM;
  if (!active) tileM = nTilesM - 1;          // clamp: still loads valid memory, no stores
  const int col0  = blockIdx.x * 64;
  const int r16   = lane & 15;
  const int khalf = (lane >> 4) << 1;        // 0 for lanes 0-15, 2 for lanes 16-31
  const float* aPtr = A + (size_t)(tileM * 16 + r16) * K;

  // cooperatively issue async loads of W[kbase..kbase+31, col0..col0+63] into wpanel[buf]
  auto issue_panel = [&](int kbase, int buf) {
#pragma unroll
    for (int i = 0; i < 4; ++i) {
      int q   = tid + i * 128;               // 512 float4 elements total
      int row = q >> 4;
      int c4  = (q & 15) << 2;
      const float* g = W + (size_t)(kbase + row) * Nout + col0 + c4;
      unsigned ldsOff = (unsigned)(uintptr_t)(&wpanel[buf][row * 64 + c4]);
      asm volatile("global_load_async_to_lds_b128 %0, %1, off"
                   :: "v"(ldsOff), "v"(g) : "memory");
    }
  };

  issue_panel(0, 0);
  asm volatile("s_wait_asynccnt 0" ::: "memory");
  __syncthreads();

  v8f acc0 = {}, acc1 = {}, acc2 = {}, acc3 = {};
  const int nChunks = K / KC;
  for (int c = 0; c < nChunks; ++c) {
    if (c + 1 < nChunks) issue_panel((c + 1) * KC, (c + 1) & 1);
    const float* wp = wpanel[c & 1];
    const int kb = c * KC;
#pragma unroll
    for (int k0 = 0; k0 < KC; k0 += 4) {
      v2f a;
      // A 16x4 f32 fragment: lanes 0-15 hold K=k0,k0+1; lanes 16-31 hold K=k0+2,k0+3
      a.x = aPtr[kb + k0 + khalf];
      a.y = aPtr[kb + k0 + khalf + 1];
      const float* w0 = wp + (k0 + khalf) * 64;
      const float* w1 = w0 + 64;
      v2f b0, b1, b2, b3;
      b0.x = w0[r16];      b0.y = w1[r16];
      b1.x = w0[16 + r16]; b1.y = w1[16 + r16];
      b2.x = w0[32 + r16]; b2.y = w1[32 + r16];
      b3.x = w0[48 + r16]; b3.y = w1[48 + r16];
      acc0 = __builtin_amdgcn_wmma_f32_16x16x4_f32(false, a, false, b0, (short)0, acc0, false, false);
      acc1 = __builtin_amdgcn_wmma_f32_16x16x4_f32(false, a, false, b1, (short)0, acc1, false, false);
      acc2 = __builtin_amdgcn_wmma_f32_16x16x4_f32(false, a, false, b2, (short)0, acc2, false, false);
      acc3 = __builtin_amdgcn_wmma_f32_16x16x4_f32(false, a, false, b3, (short)0, acc3, false, false);
    }
    asm volatile("s_wait_asynccnt 0" ::: "memory");
    __syncthreads();
  }

  if (active) {
    // C/D layout: VGPR r -> M = r (lanes 0-15) or r+8 (lanes 16-31); N = lane%16
    const int rbase = tileM * 16 + ((lane >> 4) << 3);
    v8f accs[4] = {acc0, acc1, acc2, acc3};
#pragma unroll
    for (int nt = 0; nt < 4; ++nt) {
      const int col = col0 + nt * 16 + r16;
      const float bv = bias ? bias[col] : 0.f;
#pragma unroll
      for (int r = 0; r < 8; ++r) {
        C[(size_t)(rbase + r) * Nout + col] = accs[nt][r] + bv;
      }
    }
  }
}

// ---------------- small utility kernels ----------------
__global__ void k_fill(float* p, size_t n, float v) {
  size_t i = (size_t)blockIdx.x * blockDim.x + threadIdx.x;
  if (i < n) p[i] = v;
}

__global__ void k_counts(const int* __restrict__ gid, float* counts, int n) {
  int i = blockIdx.x * blockDim.x + threadIdx.x;
  if (i < n) atomicAdd(&counts[gid[i]], 1.f);
}

// hcur[n,j] += proj_W[wid[n], j]  (the one-hot @ proj_W[:V] part of the projection)
__global__ void k_add_vocab(float* __restrict__ hc, const int* __restrict__ wid,
                            const float* __restrict__ projW, int total) {
  int i = blockIdx.x * blockDim.x + threadIdx.x;
  if (i >= total) return;
  int n = i / H, j = i - n * H;
  hc[i] += projW[(size_t)wid[n] * H + j];
}

// ro[gid[n], r, j] += x[n, j]
__global__ void k_readout_accum(const float* __restrict__ x, const int* __restrict__ gid,
                                float* __restrict__ ro, int r, int total) {
  int i = blockIdx.x * blockDim.x + threadIdx.x;
  if (i >= total) return;
  int n = i / H, j = i - n * H;
  atomicAdd(&ro[(size_t)gid[n] * (T_SEQ * H) + r * H + j], x[i]);
}

__global__ void k_readout_div(float* __restrict__ ro, const float* __restrict__ counts,
                              int total) {
  int i = blockIdx.x * blockDim.x + threadIdx.x;
  if (i >= total) return;
  int b = i / (T_SEQ * H);
  ro[i] /= fmaxf(counts[b], 1.f);
}

// ---------------- GAT edge phase ----------------
// Per-node attention halves: a_u[n,hd] = <wv[n,hd,:], att_u[hd]>, a_v likewise.
__global__ void k_node_att(const float* __restrict__ wv, const float* __restrict__ att,
                           float* __restrict__ a_u, float* __restrict__ a_v, int nTasks) {
  int gw = (blockIdx.x * blockDim.x + threadIdx.x) >> 5;
  if (gw >= nTasks) return;
  int lane = threadIdx.x & 31;
  int n = gw / HEADS, hd = gw - n * HEADS;
  const float* wrow = wv + (size_t)n * (HEADS * H) + hd * H;
  const float* au = att + hd * (2 * H);
  const float* av = au + H;
  float su = 0.f, sv = 0.f;
  for (int j = lane; j < H; j += 32) {
    float w = wrow[j];
    su += w * au[j];
    sv += w * av[j];
  }
  for (int o = 16; o; o >>= 1) {
    su += __shfl_xor(su, o, 32);
    sv += __shfl_xor(sv, o, 32);
  }
  if (lane == 0) { a_u[gw] = su; a_v[gw] = sv; }
}

// attn = leaky_relu(a_u[src] + a_v[dst]); segment max over dst via int-trick atomic max.
__global__ void k_edge_attn(const int* __restrict__ src, const int* __restrict__ dst,
                            const float* __restrict__ a_u, const float* __restrict__ a_v,
                            float* __restrict__ ebuf, float* __restrict__ m, int total) {
  int i = blockIdx.x * blockDim.x + threadIdx.x;
  if (i >= total) return;
  int e = i / HEADS, hd = i - e * HEADS;
  float v = a_u[src[e] * HEADS + hd] + a_v[dst[e] * HEADS + hd];
  v = v > 0.f ? v : NEG_SLOPE * v;
  ebuf[i] = v;
  float* addr = &m[dst[e] * HEADS + hd];
  if (v >= 0.f) atomicMax((int*)addr, __float_as_int(v));
  else          atomicMin((unsigned int*)addr, __float_as_uint(v));
}

// e = exp(attn - m[dst]); segment sum over dst.
__global__ void k_edge_exp(const int* __restrict__ dst, float* __restrict__ ebuf,
                           const float* __restrict__ m, float* __restrict__ s, int total) {
  int i = blockIdx.x * blockDim.x + threadIdx.x;
  if (i >= total) return;
  int e = i / HEADS, hd = i - e * HEADS;
  int di = dst[e] * HEADS + hd;
  float mm = m[di];
  if (mm == -INFINITY) mm = 0.f;
  float ev = expf(ebuf[i] - mm);
  ebuf[i] = ev;
  atomicAdd(&s[di], ev);
}

// agg[dst,hd,:] += (e/s[dst]) * wv[src,hd,:] ; one wave per (e,hd), float4 per lane.
__global__ void k_edge_agg(const int* __restrict__ src, const int* __restrict__ dst,
                           const float* __restrict__ ebuf, const float* __restrict__ s,
                           const float* __restrict__ wv, float* __restrict__ agg, int nTasks) {
  int gw = (blockIdx.x * blockDim.x + threadIdx.x) >> 5;
  if (gw >= nTasks) return;
  int lane = threadIdx.x & 31;
  int e = gw / HEADS, hd = gw - e * HEADS;
  int se = src[e], de = dst[e];
  float score = ebuf[gw] / s[de * HEADS + hd];
  const float4 wval =
      reinterpret_cast<const float4*>(wv + (size_t)se * (HEADS * H) + hd * H)[lane];
  float* out = agg + (size_t)de * (HEADS * H) + hd * H + lane * 4;
  atomicAdd(out + 0, score * wval.x);
  atomicAdd(out + 1, score * wval.y);
  atomicAdd(out + 2, score * wval.z);
  atomicAdd(out + 3, score * wval.w);
}

// residual + zero-in-degree mask + LayerNorm + ReLU. One wave per node (4 feats/lane).
__global__ void k_finalize(const float* __restrict__ res, const float* __restrict__ hin,
                           const float* __restrict__ s, const float* __restrict__ gamma,
                           const float* __restrict__ beta, float* __restrict__ hout, int Nn) {
  int gw = (blockIdx.x * blockDim.x + threadIdx.x) >> 5;
  if (gw >= Nn) return;
  int lane = threadIdx.x & 31;
  int n = gw;
  bool has_in = s[n * HEADS] > 0.f;
  float4 rv = has_in ? reinterpret_cast<const float4*>(res + (size_t)n * H)[lane]
                     : make_float4(0.f, 0.f, 0.f, 0.f);
  float4 hv = reinterpret_cast<const float4*>(hin + (size_t)n * H)[lane];
  float x0 = rv.x + hv.x, x1 = rv.y + hv.y, x2 = rv.z + hv.z, x3 = rv.w + hv.w;
  float sum = x0 + x1 + x2 + x3;
  float sq = x0 * x0 + x1 * x1 + x2 * x2 + x3 * x3;
  for (int o = 16; o; o >>= 1) {
    sum += __shfl_xor(sum, o, 32);
    sq  += __shfl_xor(sq, o, 32);
  }
  float mean = sum * (1.f / H);
  float var = sq * (1.f / H) - mean * mean;
  float inv = rsqrtf(var + LN_EPS);
  int j = lane * 4;
  float y0 = (x0 - mean) * inv * gamma[j + 0] + beta[j + 0];
  float y1 = (x1 - mean) * inv * gamma[j + 1] + beta[j + 1];
  float y2 = (x2 - mean) * inv * gamma[j + 2] + beta[j + 2];
  float y3 = (x3 - mean) * inv * gamma[j + 3] + beta[j + 3];
  float* out = hout + (size_t)n * H + j;
  out[0] = fmaxf(y0, 0.f);
  out[1] = fmaxf(y1, 0.f);
  out[2] = fmaxf(y2, 0.f);
  out[3] = fmaxf(y3, 0.f);
}

// ---------------- GRU (single direction, single layer) ----------------
__device__ __forceinline__ float sigmoidf_(float x) { return 1.f / (1.f + expf(-x)); }

__global__ void k_gru_dir(const float* __restrict__ x, int in_dim,
                          const float* __restrict__ Wih, const float* __restrict__ Whh,
                          const float* __restrict__ bih, const float* __restrict__ bhh,
                          float* __restrict__ ys, float* __restrict__ hT, int reverse) {
  __shared__ float xsh[2 * H];
  __shared__ float hsh[H];
  int b = blockIdx.x;
  int j = threadIdx.x;      // 0..127
  hsh[j] = 0.f;
  __syncthreads();
  for (int step = 0; step < T_SEQ; ++step) {
    int t = reverse ? (T_SEQ - 1 - step) : step;
    for (int i = j; i < in_dim; i += H)
      xsh[i] = x[((size_t)b * T_SEQ + t) * in_dim + i];
    __syncthreads();
    float gir = bih[j], giz = bih[H + j], gin = bih[2 * H + j];
    const float* wr = Wih + (size_t)j * in_dim;
    const float* wz = Wih + (size_t)(H + j) * in_dim;
    const float* wn = Wih + (size_t)(2 * H + j) * in_dim;
    for (int k = 0; k < in_dim; ++k) {
      float xv = xsh[k];
      gir += wr[k] * xv;
      giz += wz[k] * xv;
      gin += wn[k] * xv;
    }
    float ghr = bhh[j], ghz = bhh[H + j], ghn = bhh[2 * H + j];
    const float* ur = Whh + (size_t)j * H;
    const float* uz = Whh + (size_t)(H + j) * H;
    const float* un = Whh + (size_t)(2 * H + j) * H;
    for (int k = 0; k < H; ++k) {
      float hv = hsh[k];
      ghr += ur[k] * hv;
      ghz += uz[k] * hv;
      ghn += un[k] * hv;
    }
    float r = sigmoidf_(gir + ghr);
    float z = sigmoidf_(giz + ghz);
    float nv = tanhf(gin + r * ghn);
    float hnew = (1.f - z) * nv + z * hsh[j];
    __syncthreads();
    hsh[j] = hnew;
    ys[((size_t)b * T_SEQ + t) * H + j] = hnew;
    __syncthreads();
  }
  hT[(size_t)b * H + j] = hsh[j];
}

__global__ void k_concat2(const float* __restrict__ yf, const float* __restrict__ yb,
                          float* __restrict__ xcat, int total /*B*T*H*/) {
  int i = blockIdx.x * blockDim.x + threadIdx.x;
  if (i >= total) return;
  int bt = i / H, j = i - bt * H;
  xcat[(size_t)bt * (2 * H) + j] = yf[i];
  xcat[(size_t)bt * (2 * H) + H + j] = yb[i];
}

__global__ void k_mean4(const float* __restrict__ f, float* __restrict__ out, int total) {
  int i = blockIdx.x * blockDim.x + threadIdx.x;
  if (i >= total) return;
  out[i] = 0.25f * (f[i] + f[total + i] + f[2 * total + i] + f[3 * total + i]);
}

// ---------------- host-side launch sequence ----------------
extern "C" void kernel_launch(void* const* d_in, const int* in_sizes, int n_in,
                              void* d_out, int out_size, void* d_ws, size_t ws_size,
                              hipStream_t stream) {
  (void)in_sizes; (void)n_in; (void)out_size; (void)ws_size;
  const float* h     = (const float*)d_in[0];
  const int*   wid   = (const int*)d_in[1];
  const int*   src   = (const int*)d_in[2];
  const int*   dst   = (const int*)d_in[3];
  const int*   gid   = (const int*)d_in[4];
  const float* projW = (const float*)d_in[5];
  const float* projb = (const float*)d_in[6];
  const float* Wn    = (const float*)d_in[7];   // [L, H, 3H]
  const float* bn    = (const float*)d_in[8];   // [L, 3H]
  const float* attw  = (const float*)d_in[9];   // [L, 3, 2H]
  const float* Ws    = (const float*)d_in[10];  // [L, 3H, H]
  const float* bcv   = (const float*)d_in[11];  // [L, H]
  const float* gam   = (const float*)d_in[12];
  const float* bet   = (const float*)d_in[13];
  const float* Wih0  = (const float*)d_in[14];  // [2, 3H, H]
  const float* Whh0  = (const float*)d_in[15];  // [2, 3H, H]
  const float* bih0  = (const float*)d_in[16];  // [2, 3H]
  const float* bhh0  = (const float*)d_in[17];
  const float* Wih1  = (const float*)d_in[18];  // [2, 3H, 2H]
  const float* Whh1  = (const float*)d_in[19];  // [2, 3H, H]
  const float* bih1  = (const float*)d_in[20];
  const float* bhh1  = (const float*)d_in[21];

  // workspace layout (floats, 16B-aligned chunks)
  float* ws = (float*)d_ws;
  size_t off = 0;
  auto wsalloc = [&](size_t n) {
    float* p = ws + off;
    off += (n + 3) & ~(size_t)3;
    return p;
  };
  float* hcur   = wsalloc((size_t)N_NODES * H);
  float* hnext  = wsalloc((size_t)N_NODES * H);
  float* wv     = wsalloc((size_t)N_NODES * HEADS * H);
  float* agg    = wsalloc((size_t)N_NODES * HEADS * H);
  float* res    = wsalloc((size_t)N_NODES * H);
  float* a_u    = wsalloc((size_t)N_NODES * HEADS);
  float* a_v    = wsalloc((size_t)N_NODES * HEADS);
  float* mbuf   = wsalloc((size_t)N_NODES * HEADS);
  float* sbuf   = wsalloc((size_t)N_NODES * HEADS);
  float* ebuf   = wsalloc((size_t)N_EDGES * HEADS);
  float* ro     = wsalloc((size_t)BATCH * T_SEQ * H);
  float* counts = wsalloc(BATCH);
  float* yf     = wsalloc((size_t)BATCH * T_SEQ * H);
  float* yb     = wsalloc((size_t)BATCH * T_SEQ * H);
  float* xcat   = wsalloc((size_t)BATCH * T_SEQ * 2 * H);
  float* finals = wsalloc((size_t)4 * BATCH * H);

  auto fill = [&](float* p, size_t n, float v) {
    k_fill<<<(unsigned)((n + 255) / 256), 256, 0, stream>>>(p, n, v);
  };
  const int mTileBlocks = (N_NODES / 16 + 3) / 4;   // ceil(1250/4) = 313

  // readout accumulators
  fill(ro, (size_t)BATCH * T_SEQ * H, 0.f);
  fill(counts, BATCH, 0.f);
  k_counts<<<(N_NODES + 255) / 256, 256, 0, stream>>>(gid, counts, N_NODES);

  // projection: hcur = h @ proj_W[V:] + proj_b + proj_W[wid]
  k_gemm_wmma<<<dim3(H / 64, mTileBlocks), 128, 0, stream>>>(
      h, projW + (size_t)VOCAB * H, projb, hcur, N_NODES, H, H);
  k_add_vocab<<<(N_NODES * H + 255) / 256, 256, 0, stream>>>(hcur, wid, projW, N_NODES * H);
  k_readout_accum<<<(N_NODES * H + 255) / 256, 256, 0, stream>>>(hcur, gid, ro, 0, N_NODES * H);

  // GAT layers
  for (int l = 0; l < LAYERS; ++l) {
    const float* Wnl  = Wn + (size_t)l * H * HEADS * H;
    const float* bnl  = bn + (size_t)l * HEADS * H;
    const float* attl = attw + (size_t)l * HEADS * 2 * H;
    const float* Wsl  = Ws + (size_t)l * HEADS * H * H;
    const float* bl   = bcv + (size_t)l * H;
    const float* gl   = gam + (size_t)l * H;
    const float* btl  = bet + (size_t)l * H;

    // wv = hcur @ Wn + bn   [N,128] x [128,384]
    k_gemm_wmma<<<dim3((HEADS * H) / 64, mTileBlocks), 128, 0, stream>>>(
        hcur, Wnl, bnl, wv, N_NODES, H, HEADS * H);

    // per-node attention projections
    k_node_att<<<(N_NODES * HEADS) / 4, 128, 0, stream>>>(wv, attl, a_u, a_v,
                                                          N_NODES * HEADS);

    // reset segment accumulators
    fill(agg, (size_t)N_NODES * HEADS * H, 0.f);
    fill(sbuf, (size_t)N_NODES * HEADS, 0.f);
    fill(mbuf, (size_t)N_NODES * HEADS, -INFINITY);

    // edge phase: logits+max, exp+sum, weighted aggregation
    k_edge_attn<<<(N_EDGES * HEADS + 255) / 256, 256, 0, stream>>>(
        src, dst, a_u, a_v, ebuf, mbuf, N_EDGES * HEADS);
    k_edge_exp<<<(N_EDGES * HEADS + 255) / 256, 256, 0, stream>>>(
        dst, ebuf, mbuf, sbuf, N_EDGES * HEADS);
    k_edge_agg<<<(N_EDGES * HEADS) / 4, 128, 0, stream>>>(
        src, dst, ebuf, sbuf, wv, agg, N_EDGES * HEADS);

    // res = agg @ Ws + b    [N,384] x [384,128]
    k_gemm_wmma<<<dim3(H / 64, mTileBlocks), 128, 0, stream>>>(
        agg, Wsl, bl, res, N_NODES, HEADS * H, H);

    // residual + mask + LayerNorm + ReLU
    k_finalize<<<N_NODES / 4, 128, 0, stream>>>(res, hcur, sbuf, gl, btl, hnext, N_NODES);
    std::swap(hcur, hnext);

    k_readout_accum<<<(N_NODES * H + 255) / 256, 256, 0, stream>>>(hcur, gid, ro, l + 1,
                                                                   N_NODES * H);
  }

  // mean readouts
  k_readout_div<<<(BATCH * T_SEQ * H + 255) / 256, 256, 0, stream>>>(
      ro, counts, BATCH * T_SEQ * H);

  // GRU layer 0 (bidirectional, input = readout sequence [B,T,H])
  k_gru_dir<<<BATCH, H, 0, stream>>>(ro, H, Wih0, Whh0, bih0, bhh0, yf, finals, 0);
  k_gru_dir<<<BATCH, H, 0, stream>>>(ro, H, Wih0 + (size_t)3 * H * H,
                                     Whh0 + (size_t)3 * H * H, bih0 + 3 * H, bhh0 + 3 * H,
                                     yb, finals + (size_t)BATCH * H, 1);
  k_concat2<<<(BATCH * T_SEQ * H + 255) / 256, 256, 0, stream>>>(yf, yb, xcat,
                                                                 BATCH * T_SEQ * H);
  // GRU layer 1 (input = [B,T,2H])
  k_gru_dir<<<BATCH, H, 0, stream>>>(xcat, 2 * H, Wih1, Whh1, bih1, bhh1, yf,
                                     finals + (size_t)2 * BATCH * H, 0);
  k_gru_dir<<<BATCH, H, 0, stream>>>(xcat, 2 * H, Wih1 + (size_t)3 * H * 2 * H,
                                     Whh1 + (size_t)3 * H * H, bih1 + 3 * H, bhh1 + 3 * H,
                                     yb, finals + (size_t)3 * BATCH * H, 1);

  // mean of 4 final hidden states -> d_out [B,H]
  k_mean4<<<(BATCH * H + 255) / 256, 256, 0, stream>>>(finals, (float*)d_out, BATCH * H);
}